// OctreeDepad_24146306138656
// MI455X (gfx1250) — compile-verified
//
#include <hip/hip_runtime.h>
#include <hip/hip_bf16.h>

// octree_depad: out[c][i] = in[c][kept_idx[i]]  (C=32 channels)
// Pure gather, 0 FLOPs -> bandwidth bound (~390 MB @ 23.3 TB/s ~= 17 us).
// Strategy: 4 kept indices per thread, per-channel 4 scalar gather loads +
// one b128 vector store, all with TH=NT (streaming, zero reuse) cache hints.

typedef __attribute__((ext_vector_type(4))) float v4f;

#ifndef OD_CHANNELS
#define OD_CHANNELS 32
#endif

__global__ __launch_bounds__(256) void OctreeDepad_gather_kernel(
    const float* __restrict__ in,        // [C, n_pad]
    const long long* __restrict__ idx,   // [n_kept] sorted int64
    float* __restrict__ out,             // [C, n_kept]
    int n_kept, int n_pad) {
  const int i0 = (blockIdx.x * blockDim.x + threadIdx.x) * 4;
  if (i0 >= n_kept) return;

  if (i0 + 4 <= n_kept) {
    // Truncate to 32-bit offsets: n_pad = 2^21 fits easily; lets the backend
    // use saddr(64b, scalar per-channel base) + 32-bit vgpr offset (GVS mode).
    const unsigned int k0 = (unsigned int)__builtin_nontemporal_load(&idx[i0 + 0]);
    const unsigned int k1 = (unsigned int)__builtin_nontemporal_load(&idx[i0 + 1]);
    const unsigned int k2 = (unsigned int)__builtin_nontemporal_load(&idx[i0 + 2]);
    const unsigned int k3 = (unsigned int)__builtin_nontemporal_load(&idx[i0 + 3]);

#pragma unroll 4
    for (int c = 0; c < OD_CHANNELS; ++c) {
      const float* __restrict__ src = in + (size_t)c * (size_t)n_pad;
      v4f v;
      v.x = __builtin_nontemporal_load(src + k0);
      v.y = __builtin_nontemporal_load(src + k1);
      v.z = __builtin_nontemporal_load(src + k2);
      v.w = __builtin_nontemporal_load(src + k3);
      v4f* __restrict__ dst =
          (v4f*)(out + (size_t)c * (size_t)n_kept + (size_t)i0);
      __builtin_nontemporal_store(v, dst);
    }
  } else {
    // Scalar tail (not hit for n_kept = 2^20, kept for robustness).
    for (int i = i0; i < n_kept; ++i) {
      const unsigned int k = (unsigned int)idx[i];
      for (int c = 0; c < OD_CHANNELS; ++c) {
        out[(size_t)c * (size_t)n_kept + (size_t)i] =
            in[(size_t)c * (size_t)n_pad + (size_t)k];
      }
    }
  }
}

extern "C" void kernel_launch(void* const* d_in, const int* in_sizes, int n_in,
                              void* d_out, int out_size, void* d_ws, size_t ws_size,
                              hipStream_t stream) {
  (void)n_in; (void)d_ws; (void)ws_size; (void)out_size;
  const float* in = (const float*)d_in[0];             // float32 [1,C,N_PAD,1]
  const long long* idx = (const long long*)d_in[1];    // int64   [N_KEPT]
  float* out = (float*)d_out;                          // float32 [1,C,N_KEPT,1]

  const int n_kept = in_sizes[1];                      // 1048576
  const int n_pad = in_sizes[0] / OD_CHANNELS;         // 2097152

  const int threads = 256;
  const int per_block = threads * 4;                   // 4 indices per thread
  const int blocks = (n_kept + per_block - 1) / per_block;

  OctreeDepad_gather_kernel<<<blocks, threads, 0, stream>>>(in, idx, out,
                                                            n_kept, n_pad);
}